// GCN3norm_40956808135024
// MI455X (gfx1250) — compile-verified
//
#include <hip/hip_runtime.h>
#include <hip/hip_bf16.h>

// ---------------------------------------------------------------------------
// GCN: h1 = relu(seg_sum(X@W1) + b1); h2 = relu(seg_sum(h1@W2) + b2);
//      g = groupnorm(h2); out = log_softmax(seg_sum(g@W3) + b3)
// MI455X: scatter-add bound (~2 GB @ 23.3 TB/s); GEMMs use native-fp32
// V_WMMA_F32_16X16X4_F32 (exact fp32 products). Wave32; weights staged in LDS
// (320 KB/WGP) packed as (k,k+1) float2 pairs so B feeds straight into the
// WMMA lane layout with single ds_load_b64 per tile per K-step.
// ---------------------------------------------------------------------------

typedef float v2f __attribute__((ext_vector_type(2)));
typedef float v8f __attribute__((ext_vector_type(8)));

#define NFEAT   128
#define NHID    96
#define NCLASS  40
#define GN_EPS  1e-5f

// C[M x ncols] = A[M x K] * B[K x ncols], row-major.
// Block = 128 threads (4 waves); wave w computes M-tile (blockIdx.x*4 + w)
// across the whole padded N-strip (NT tiles of 16). W is staged in LDS,
// zero-padded to NT*16 columns, packed as lds2[k/2][col] = {B[k][col], B[k+1][col]}.
// ISA 7.12.2 (V_WMMA_F32_16X16X4_F32, wave32):
//   A 16x4 : lanes 0-15: M=lane, v0=K0 v1=K1 ; lanes 16-31: v0=K2 v1=K3
//   B 4x16 : lanes 0-15: N=lane, v0=K0 v1=K1 ; lanes 16-31: v0=K2 v1=K3
//   C 16x16: vgpr r: lanes 0-15 -> M=r, lanes 16-31 -> M=r+8 ; N = lane&15
template<int NT, int KMAX>
__global__ __launch_bounds__(128) void wmma_gemm_strip_kernel(
    const float* __restrict__ A, const float* __restrict__ Bw,
    float* __restrict__ C, int K, int ncols, int ntm)
{
    constexpr int NCP = NT * 16;            // padded column count
    __shared__ float lds[KMAX * NCP];       // packed pair layout (float2 view)

    const int tid = threadIdx.x;

    // Cooperative fill: lds2[(k>>1)*NCP + col] = {B[k][col], B[k+1][col]},
    // zero for col >= ncols (so the inner loop needs no guards).
    for (int i = tid; i < K * NCP; i += 128) {
        int k   = i / NCP;
        int col = i - k * NCP;
        float v = (col < ncols) ? Bw[(size_t)k * ncols + col] : 0.0f;
        lds[(k >> 1) * (NCP * 2) + col * 2 + (k & 1)] = v;
    }
    __syncthreads();

    const int wave  = tid >> 5;
    const int mtile = blockIdx.x * 4 + wave;
    if (mtile >= ntm) return;               // wave-uniform: EXEC stays all-1s

    const int lane  = tid & 31;
    const int coll  = lane & 15;
    const int mrow  = mtile * 16 + coll;
    const int khalf = (lane >> 4) * 2;      // 0 (lanes 0-15) / 2 (lanes 16-31)

    const float2* __restrict__ lds2 = reinterpret_cast<const float2*>(lds);

    v8f acc[NT] = {};
    for (int k = 0; k < K; k += 4) {
        const int kk = k + khalf;
        float2 av = *reinterpret_cast<const float2*>(A + (size_t)mrow * K + kk);
        v2f a; a.x = av.x; a.y = av.y;
        const float2* bp = lds2 + (kk >> 1) * NCP + coll;
        #pragma unroll
        for (int t = 0; t < NT; ++t) {
            float2 bv = bp[t * 16];         // ds_load_b64
            v2f b; b.x = bv.x; b.y = bv.y;
            acc[t] = __builtin_amdgcn_wmma_f32_16x16x4_f32(
                false, a, false, b, (short)0, acc[t], false, false);
        }
    }

    const int rbase = mtile * 16 + (lane >> 4) * 8;
    #pragma unroll
    for (int t = 0; t < NT; ++t) {
        int col = t * 16 + coll;
        if (col < ncols) {
            #pragma unroll
            for (int r = 0; r < 8; ++r)
                C[(size_t)(rbase + r) * ncols + col] = acc[t][r];
        }
    }
}

// agg[tgt[e], :] += h[src[e], :] * w[e]; one thread per (edge, 4-channel vec).
__global__ void scatter_kernel(const float* __restrict__ h,
                               const int* __restrict__ src,
                               const int* __restrict__ tgt,
                               const float* __restrict__ w,
                               float* __restrict__ agg,
                               int C, int vpe, int total)
{
    int idx = blockIdx.x * blockDim.x + threadIdx.x;
    if (idx >= total) return;
    int e = idx / vpe;
    int c = (idx - e * vpe) * 4;
    int s = src[e];
    int t = tgt[e];
    float ww = w[e];
    float4 m = *reinterpret_cast<const float4*>(h + (size_t)s * C + c);
    float* ap = agg + (size_t)t * C + c;
    atomicAdd(ap + 0, m.x * ww);   // -> global_atomic_add_f32
    atomicAdd(ap + 1, m.y * ww);
    atomicAdd(ap + 2, m.z * ww);
    atomicAdd(ap + 3, m.w * ww);
}

__global__ void bias_relu_kernel(float* __restrict__ x,
                                 const float* __restrict__ b,
                                 int C, int total)
{
    int idx = blockIdx.x * blockDim.x + threadIdx.x;
    if (idx >= total) return;
    int c = idx % C;
    x[idx] = fmaxf(x[idx] + b[c], 0.0f);
}

// GroupNorm over [N,96], 32 groups of 3 channels; one thread per (row, group).
__global__ void groupnorm_kernel(float* __restrict__ x,
                                 const float* __restrict__ gamma,
                                 const float* __restrict__ beta,
                                 int total /* N*32 */)
{
    int idx = blockIdx.x * blockDim.x + threadIdx.x;
    if (idx >= total) return;
    int row = idx >> 5;
    int g   = idx & 31;
    size_t base = (size_t)row * NHID + g * 3;
    float x0 = x[base], x1 = x[base + 1], x2 = x[base + 2];
    float mu = (x0 + x1 + x2) * (1.0f / 3.0f);
    float d0 = x0 - mu, d1 = x1 - mu, d2 = x2 - mu;
    float var = (d0 * d0 + d1 * d1 + d2 * d2) * (1.0f / 3.0f);
    float inv = rsqrtf(var + GN_EPS);
    int cg = g * 3;
    x[base]     = d0 * inv * gamma[cg]     + beta[cg];
    x[base + 1] = d1 * inv * gamma[cg + 1] + beta[cg + 1];
    x[base + 2] = d2 * inv * gamma[cg + 2] + beta[cg + 2];
}

// out[row,:] = log_softmax(out[row,:] + b3); one thread per row (40 classes).
__global__ void bias_logsoftmax_kernel(float* __restrict__ out,
                                       const float* __restrict__ b,
                                       int nrows)
{
    int row = blockIdx.x * blockDim.x + threadIdx.x;
    if (row >= nrows) return;
    float v[NCLASS];
    float m = -3.402823466e+38f;
    #pragma unroll
    for (int c = 0; c < NCLASS; ++c) {
        v[c] = out[(size_t)row * NCLASS + c] + b[c];
        m = fmaxf(m, v[c]);
    }
    float s = 0.0f;
    #pragma unroll
    for (int c = 0; c < NCLASS; ++c) s += __expf(v[c] - m);
    float ls = __logf(s);
    #pragma unroll
    for (int c = 0; c < NCLASS; ++c)
        out[(size_t)row * NCLASS + c] = v[c] - m - ls;
}

extern "C" void kernel_launch(void* const* d_in, const int* in_sizes, int n_in,
                              void* d_out, int out_size, void* d_ws, size_t ws_size,
                              hipStream_t stream)
{
    const float* x     = (const float*)d_in[0];
    const int*   src   = (const int*)  d_in[1];
    const int*   tgt   = (const int*)  d_in[2];
    const float* w     = (const float*)d_in[3];
    const float* W1    = (const float*)d_in[4];
    const float* b1    = (const float*)d_in[5];
    const float* W2    = (const float*)d_in[6];
    const float* b2    = (const float*)d_in[7];
    const float* W3    = (const float*)d_in[8];
    const float* b3    = (const float*)d_in[9];
    const float* gamma = (const float*)d_in[10];
    const float* beta  = (const float*)d_in[11];
    float* out = (float*)d_out;

    const int N   = in_sizes[0] / NFEAT;   // 50000 (multiple of 16)
    const int E   = in_sizes[1];           // 800000
    const int ntm = N / 16;                // 3125 M-tiles
    const int gblk = (ntm + 3) / 4;        // 4 waves (M-tiles) per block

    float* bufA = (float*)d_ws;                      // N x 96 (GEMM out)
    float* bufB = bufA + (size_t)N * NHID;           // N x 96 (agg / activ)

    const int TPB = 256;

    // ---- Layer 1: t1 = X @ W1 -> bufA ----
    wmma_gemm_strip_kernel<6, NFEAT><<<gblk, 128, 0, stream>>>(
        x, W1, bufA, NFEAT, NHID, ntm);
    hipMemsetAsync(bufB, 0, (size_t)N * NHID * sizeof(float), stream);
    {
        int vpe = NHID / 4, total = E * vpe;
        scatter_kernel<<<(total + TPB - 1) / TPB, TPB, 0, stream>>>(
            bufA, src, tgt, w, bufB, NHID, vpe, total);
        int t2 = N * NHID;
        bias_relu_kernel<<<(t2 + TPB - 1) / TPB, TPB, 0, stream>>>(bufB, b1, NHID, t2);
    }

    // ---- Layer 2: t2 = h1 @ W2 -> bufA ----
    wmma_gemm_strip_kernel<6, NFEAT><<<gblk, 128, 0, stream>>>(
        bufB, W2, bufA, NHID, NHID, ntm);
    hipMemsetAsync(bufB, 0, (size_t)N * NHID * sizeof(float), stream);
    {
        int vpe = NHID / 4, total = E * vpe;
        scatter_kernel<<<(total + TPB - 1) / TPB, TPB, 0, stream>>>(
            bufA, src, tgt, w, bufB, NHID, vpe, total);
        int t2 = N * NHID;
        bias_relu_kernel<<<(t2 + TPB - 1) / TPB, TPB, 0, stream>>>(bufB, b2, NHID, t2);
    }

    // ---- GroupNorm (in place on bufB) ----
    {
        int total = N * 32;
        groupnorm_kernel<<<(total + TPB - 1) / TPB, TPB, 0, stream>>>(
            bufB, gamma, beta, total);
    }

    // ---- Layer 3: t3 = g @ W3 -> bufA (N x 40) ----
    wmma_gemm_strip_kernel<3, NHID><<<gblk, 128, 0, stream>>>(
        bufB, W3, bufA, NHID, NCLASS, ntm);
    hipMemsetAsync(out, 0, (size_t)N * NCLASS * sizeof(float), stream);
    {
        int vpe = NCLASS / 4, total = E * vpe;
        scatter_kernel<<<(total + TPB - 1) / TPB, TPB, 0, stream>>>(
            bufA, src, tgt, w, out, NCLASS, vpe, total);
    }

    // ---- bias + log_softmax (in place on out) ----
    bias_logsoftmax_kernel<<<(N + TPB - 1) / TPB, TPB, 0, stream>>>(out, b3, N);
}